// Histogram_28905129902695
// MI455X (gfx1250) — compile-verified
//
#include <hip/hip_runtime.h>
#include <stdint.h>

// Soft local histogram, R=5, bandwidth=0.5, on 96 planes of 256x256 fp32.
// out(y,x) = sum_{5x5} max(0, 1 - 2|v - c|)  for 2<=y,x<=253, else 0
//          = 25 - 2 * sum_{5x5} min(|v - c|, 0.5)
//
// VALU-bound stencil: 3 VALU ops/tap (sub, min-with-|src|, add).
// Data staged via CDNA5 async global->LDS B128 copies (ASYNCcnt path).

#define TILE_W 64
#define TILE_H 32
#define LDS_W  72   // TILE_W + 8: halo widened to keep every B128 load 16B-aligned
#define LDS_H  36   // TILE_H + 4
#define PLANE  65536

__global__ __launch_bounds__(256) void soft_hist_kernel(const float* __restrict__ in,
                                                        float* __restrict__ out)
{
    __shared__ __align__(16) float sm[LDS_H * LDS_W];

    const int tid        = threadIdx.x;
    const int bx         = blockIdx.x * TILE_W;
    const int by         = blockIdx.y * TILE_H;
    const int plane_base = blockIdx.z * PLANE;

    // Low 32 bits of the generic pointer to a __shared__ object == LDS byte offset.
    const uint32_t lds_base = (uint32_t)(uintptr_t)(&sm[0]);

    // ---- Stage haloed tile (36 rows x 18 vec4) with async global->LDS B128 ----
    // 648 copies, 3 per thread; tail lanes clamp to a duplicate copy so EXEC
    // stays all-ones around the async ops.
#pragma unroll
    for (int k = 0; k < 3; ++k) {
        int id = tid + k * 256;
        if (id > 647) id = 647;
        const int row = id / 18;
        const int col = id - row * 18;

        int gy = by - 2 + row;               // clamp rows into the plane
        gy = gy < 0 ? 0 : (gy > 255 ? 255 : gy);
        int gx = bx - 4 + col * 4;           // clamp vec4 start into the plane
        gx = gx < 0 ? 0 : (gx > 252 ? 252 : gx);
        // Clamped halo values only influence outputs that are forced to zero.

        const int      goff  = (plane_base + gy * 256 + gx) * 4;                // bytes
        const uint32_t laddr = lds_base + (uint32_t)(row * LDS_W + col * 4) * 4u;

        asm volatile("global_load_async_to_lds_b128 %0, %1, %2"
                     :
                     : "v"(laddr), "v"(goff), "s"(in)
                     : "memory");
    }
    asm volatile("s_wait_asynccnt 0x0" ::: "memory");
    __syncthreads();

    // ---- Compute: each thread does 8 consecutive output rows at one column ----
    const int  tx  = tid & 63;
    const int  oy0 = (tid >> 6) * 8;
    const int  gx  = bx + tx;
    const bool xok = (gx >= 2) && (gx <= 253);

#pragma unroll
    for (int r = 0; r < 8; ++r) {
        const int   oy = oy0 + r;
        const float c  = sm[(oy + 2) * LDS_W + tx + 4];
        float S = 0.0f;
#pragma unroll
        for (int di = 0; di < 5; ++di) {
            const float* rp = &sm[(oy + di) * LDS_W + tx + 2];
#pragma unroll
            for (int dj = 0; dj < 5; ++dj) {
                const float d = rp[dj] - c;          // v_sub_f32
                S += fminf(fabsf(d), 0.5f);          // v_min_f32 |d|,0.5 ; v_add_f32
            }
        }
        float res = fmaf(S, -2.0f, 25.0f);

        const int gy = by + oy;
        if (!(xok && gy >= 2 && gy <= 253)) res = 0.0f;   // zero-padded border
        out[plane_base + gy * 256 + gx] = res;
    }
}

extern "C" void kernel_launch(void* const* d_in, const int* in_sizes, int n_in,
                              void* d_out, int out_size, void* d_ws, size_t ws_size,
                              hipStream_t stream)
{
    const float* in  = (const float*)d_in[0];
    float*       out = (float*)d_out;
    const int planes = in_sizes[0] >> 16;          // elements / (256*256) = 96
    dim3 grid(256 / TILE_W, 256 / TILE_H, planes);
    soft_hist_kernel<<<grid, 256, 0, stream>>>(in, out);
}